// GNNModel_17635135718115
// MI455X (gfx1250) — compile-verified
//
#include <hip/hip_runtime.h>
#include <math.h>

#define N_NODES 100000
#define N_EDGES 1600000
#define E_TOT   (N_EDGES + N_NODES)   // edges + self loops
#define D 64
#define M_TILES (N_NODES / 16)        // 100000 = 6250*16 exactly

typedef float v2f __attribute__((ext_vector_type(2)));
typedef float v8f __attribute__((ext_vector_type(8)));
typedef unsigned int u32x4 __attribute__((ext_vector_type(4)));
typedef int i32x4 __attribute__((ext_vector_type(4)));
typedef int i32x8 __attribute__((ext_vector_type(8)));

__device__ __forceinline__ void atomicAddF(float* p, float v) {
    // hardware f32 atomic add, no-return form (STOREcnt); resolves in L2
    asm volatile("global_atomic_add_f32 %0, %1, off" : : "v"(p), "v"(v) : "memory");
}

// monotonic float -> uint mapping so integer atomicMax implements float max
__device__ __forceinline__ unsigned fmap(float f) {
    unsigned b = __float_as_uint(f);
    return (b & 0x80000000u) ? ~b : (b | 0x80000000u);
}
__device__ __forceinline__ float funmap(unsigned m) {
    unsigned b = (m & 0x80000000u) ? (m & 0x7FFFFFFFu) : ~m;
    return __uint_as_float(b);
}

// ---------------- init per layer ----------------
__global__ void k_init(unsigned* __restrict__ emax, float* __restrict__ esum,
                       float* __restrict__ outacc) {
    int i = blockIdx.x * blockDim.x + threadIdx.x;
    if (i < N_NODES) { emax[i] = 0u; esum[i] = 0.0f; }
    if (i < N_NODES * D) outacc[i] = 0.0f;
}

// ---------------- layer 0 GEMM: [N,3] x [3,64] ----------------
__global__ void k_gemm3(const float* __restrict__ x, const float* __restrict__ W,
                        float* __restrict__ h) {
    int i = blockIdx.x * blockDim.x + threadIdx.x;
    if (i >= N_NODES * D) return;
    int n = i >> 6, f = i & 63;
    float acc = x[n * 3 + 0] * W[0 * D + f]
              + x[n * 3 + 1] * W[1 * D + f]
              + x[n * 3 + 2] * W[2 * D + f];
    h[i] = acc;
}

// ---------------- 64x64 GEMM via fp32 WMMA + TDM-staged weights ----------------
// W (16 KB) is DMA'd into LDS once per workgroup by the Tensor Data Mover;
// one wave per 16x64 output tile; K looped in steps of 4 (V_WMMA_F32_16X16X4_F32)
__global__ void k_gemm64_wmma(const float* __restrict__ in,  // [N,64]
                              const float* __restrict__ W,   // [64,64]
                              float* __restrict__ h) {       // [N,64]
    __shared__ float sW[D * D];   // 16 KB

    if (threadIdx.x == 0) {
        unsigned lds = (unsigned)(size_t)(void*)sW;     // low 32 bits = LDS offset
        unsigned long long ga = (unsigned long long)(size_t)W;
        u32x4 g0;
        g0[0] = 1u;                                     // count=1, user descriptor
        g0[1] = lds;                                    // lds_addr (bytes)
        g0[2] = (unsigned)(ga & 0xFFFFFFFFu);           // global_addr[31:0]
        g0[3] = (unsigned)((ga >> 32) & 0x01FFFFFFu)    // global_addr[56:32]
              | (2u << 30);                             // type=2 ("image")
        i32x8 g1;
        g1[0] = 0x00020000;        // workgroup_mask=0, data_size=2 (4 bytes)
        g1[1] = (int)(64u << 16);  // tensor_dim0 = 64 elements (bits 79:48 low half)
        g1[2] = (int)(64u << 16);  // tensor_dim1 = 64 (bits 111:80 low half)
        g1[3] = (int)(64u << 16);  // tile_dim0   = 64 (bits 127:112)
        g1[4] = 64;                // tile_dim1   = 64 (bits 143:128), tile_dim2=0
        g1[5] = 64;                // tensor_dim0_stride = 64 (bits 207:160 low)
        g1[6] = 0;                 // dim0_stride hi, dim1_stride (unused for 2D)
        g1[7] = 0;
        i32x4 z4 = {0, 0, 0, 0};
        i32x8 z8 = {0, 0, 0, 0, 0, 0, 0, 0};
        __builtin_amdgcn_tensor_load_to_lds(g0, g1, z4, z4, z8, 0);
        __builtin_amdgcn_s_wait_tensorcnt(0);           // TENSORcnt == 0
    }
    __syncthreads();

    int gtid  = blockIdx.x * blockDim.x + threadIdx.x;
    int mtile = gtid >> 5;
    int lane  = threadIdx.x & 31;
    if (mtile >= M_TILES) return;            // whole-wave uniform exit

    int row   = mtile * 16 + (lane & 15);    // A: M index from lane
    int khalf = (lane >> 4) * 2;             // lanes 16-31 hold K+2,K+3
    int col   = lane & 15;                   // B: N index from lane

    v8f c0 = {}, c1 = {}, c2 = {}, c3 = {};
    const float* arow = in + row * D;

    #pragma unroll
    for (int k = 0; k < D; k += 4) {
        v2f a;
        a.x = arow[k + khalf];
        a.y = arow[k + khalf + 1];
        int r0 = (k + khalf) * D + col;      // B fragments from LDS
        int r1 = r0 + D;
        v2f b0, b1, b2, b3;
        b0.x = sW[r0 +  0]; b0.y = sW[r1 +  0];
        b1.x = sW[r0 + 16]; b1.y = sW[r1 + 16];
        b2.x = sW[r0 + 32]; b2.y = sW[r1 + 32];
        b3.x = sW[r0 + 48]; b3.y = sW[r1 + 48];
        c0 = __builtin_amdgcn_wmma_f32_16x16x4_f32(false, a, false, b0, (short)0, c0, false, false);
        c1 = __builtin_amdgcn_wmma_f32_16x16x4_f32(false, a, false, b1, (short)0, c1, false, false);
        c2 = __builtin_amdgcn_wmma_f32_16x16x4_f32(false, a, false, b2, (short)0, c2, false, false);
        c3 = __builtin_amdgcn_wmma_f32_16x16x4_f32(false, a, false, b3, (short)0, c3, false, false);
    }

    // D layout: vgpr v -> M = v (+8 for lanes 16-31), N = lane&15
    int rbase = mtile * 16 + ((lane >> 4) ? 8 : 0);
    #pragma unroll
    for (int v = 0; v < 8; ++v) {
        float* o = h + (rbase + v) * D + col;
        o[ 0] = c0[v];
        o[16] = c1[v];
        o[32] = c2[v];
        o[48] = c3[v];
    }
}

// ---------------- per-node attention logits ----------------
// one wave per node; 64 features -> 2 per lane; wave32 shuffle reduction
__global__ void k_alpha(const float* __restrict__ h, const float* __restrict__ a_s,
                        const float* __restrict__ a_d, float* __restrict__ asrc,
                        float* __restrict__ adst) {
    int gtid = blockIdx.x * blockDim.x + threadIdx.x;
    int node = gtid >> 5;
    int lane = threadIdx.x & 31;
    if (node >= N_NODES) return;
    float h0 = h[node * D + lane];
    float h1 = h[node * D + 32 + lane];
    float s = h0 * a_s[lane] + h1 * a_s[32 + lane];
    float d = h0 * a_d[lane] + h1 * a_d[32 + lane];
    #pragma unroll
    for (int m = 16; m >= 1; m >>= 1) {
        s += __shfl_xor(s, m, 32);
        d += __shfl_xor(d, m, 32);
    }
    if (lane == 0) { asrc[node] = s; adst[node] = d; }
}

// ---------------- edge pass 1: segment max over dst ----------------
__global__ void k_edge_max(const int* __restrict__ esrc, const int* __restrict__ edst,
                           const float* __restrict__ asrc, const float* __restrict__ adst,
                           unsigned* __restrict__ emax) {
    int i = blockIdx.x * blockDim.x + threadIdx.x;
    if (i >= E_TOT) return;
    int s = (i < N_EDGES) ? esrc[i] : (i - N_EDGES);
    int d = (i < N_EDGES) ? edst[i] : (i - N_EDGES);
    float e = asrc[s] + adst[d];
    e = (e > 0.0f) ? e : 0.2f * e;           // GAT internal LeakyReLU
    atomicMax(&emax[d], fmap(e));
}

// ---------------- edge pass 2: exp + segment sum ----------------
__global__ void k_edge_sum(const int* __restrict__ esrc, const int* __restrict__ edst,
                           const float* __restrict__ asrc, const float* __restrict__ adst,
                           const unsigned* __restrict__ emax, float* __restrict__ esum,
                           float* __restrict__ pbuf) {
    int i = blockIdx.x * blockDim.x + threadIdx.x;
    if (i >= E_TOT) return;
    int s = (i < N_EDGES) ? esrc[i] : (i - N_EDGES);
    int d = (i < N_EDGES) ? edst[i] : (i - N_EDGES);
    float e = asrc[s] + adst[d];
    e = (e > 0.0f) ? e : 0.2f * e;
    float p = __expf(e - funmap(emax[d]));
    pbuf[i] = p;
    atomicAddF(&esum[d], p);
}

// ---------------- edge pass 3: weighted aggregation ----------------
// one wave per edge; each lane moves 2 features (float2 gather, 2x f32 atomic add)
__global__ void k_aggregate(const int* __restrict__ esrc, const int* __restrict__ edst,
                            const float* __restrict__ pbuf, const float* __restrict__ esum,
                            const float* __restrict__ h, float* __restrict__ outacc) {
    int gtid = blockIdx.x * blockDim.x + threadIdx.x;
    int edge = gtid >> 5;
    int lane = threadIdx.x & 31;
    if (edge >= E_TOT) return;
    int s = (edge < N_EDGES) ? esrc[edge] : (edge - N_EDGES);
    int d = (edge < N_EDGES) ? edst[edge] : (edge - N_EDGES);
    float alpha = pbuf[edge] / (esum[d] + 1e-16f);
    const float2 v = ((const float2*)(h + s * D))[lane];
    float* o = outacc + d * D + lane * 2;
    atomicAddF(o + 0, alpha * v.x);
    atomicAddF(o + 1, alpha * v.y);
}

// ---------------- bias + (optional) activation ----------------
__global__ void k_finalize(const float* __restrict__ outacc, const float* __restrict__ bias,
                           float* __restrict__ dst, int apply_act) {
    int i = blockIdx.x * blockDim.x + threadIdx.x;
    if (i >= N_NODES * D) return;
    float v = outacc[i] + bias[i & 63];
    if (apply_act) v = (v > 0.0f) ? v : 0.01f * v;
    dst[i] = v;
}

extern "C" void kernel_launch(void* const* d_in, const int* in_sizes, int n_in,
                              void* d_out, int out_size, void* d_ws, size_t ws_size,
                              hipStream_t stream) {
    (void)in_sizes; (void)n_in; (void)out_size; (void)ws_size;

    const float* x    = (const float*)d_in[0];
    const int*   eidx = (const int*)d_in[1];         // [2, E] flattened
    const int*   esrc = eidx;
    const int*   edst = eidx + N_EDGES;

    // per-layer params: {W, a_src, a_dst, b}
    const float* W[4]  = { (const float*)d_in[2],
                           (const float*)d_in[6],
                           (const float*)d_in[6] + D * D,
                           (const float*)d_in[10] };
    const float* AS[4] = { (const float*)d_in[3],
                           (const float*)d_in[7],
                           (const float*)d_in[7] + D,
                           (const float*)d_in[11] };
    const float* AD[4] = { (const float*)d_in[4],
                           (const float*)d_in[8],
                           (const float*)d_in[8] + D,
                           (const float*)d_in[12] };
    const float* B[4]  = { (const float*)d_in[5],
                           (const float*)d_in[9],
                           (const float*)d_in[9] + D,
                           (const float*)d_in[13] };

    // workspace carve-up (~85 MB of f32)
    float*    ws    = (float*)d_ws;
    float*    bufA  = ws;                       // layer input features  [N,64]
    float*    bufH  = bufA + N_NODES * D;       // h = in @ W            [N,64]
    float*    bufO  = bufH + N_NODES * D;       // aggregation accum     [N,64]
    float*    asrc  = bufO + N_NODES * D;       // [N]
    float*    adst  = asrc + N_NODES;           // [N]
    float*    esum  = adst + N_NODES;           // [N]
    unsigned* emax  = (unsigned*)(esum + N_NODES);  // [N]
    float*    pbuf  = (float*)(emax + N_NODES);     // [E_TOT]

    const int TPB = 256;
    const int gNodeFeat = (N_NODES * D + TPB - 1) / TPB;
    const int gEdge     = (E_TOT + TPB - 1) / TPB;
    const int gEdgeWave = (E_TOT * 32 + TPB - 1) / TPB;   // one wave per edge
    const int gNodeWave = (N_NODES * 32 + TPB - 1) / TPB; // one wave per node
    const int gGemm     = (M_TILES + (TPB / 32) - 1) / (TPB / 32);

    for (int L = 0; L < 4; ++L) {
        k_init<<<gNodeFeat, TPB, 0, stream>>>(emax, esum, bufO);

        if (L == 0) k_gemm3<<<gNodeFeat, TPB, 0, stream>>>(x, W[0], bufH);
        else        k_gemm64_wmma<<<gGemm, TPB, 0, stream>>>(bufA, W[L], bufH);

        k_alpha<<<gNodeWave, TPB, 0, stream>>>(bufH, AS[L], AD[L], asrc, adst);
        k_edge_max<<<gEdge, TPB, 0, stream>>>(esrc, edst, asrc, adst, emax);
        k_edge_sum<<<gEdge, TPB, 0, stream>>>(esrc, edst, asrc, adst, emax, esum, pbuf);
        k_aggregate<<<gEdgeWave, TPB, 0, stream>>>(esrc, edst, pbuf, esum, bufH, bufO);

        float* dst = (L == 3) ? (float*)d_out : bufA;
        k_finalize<<<gNodeFeat, TPB, 0, stream>>>(bufO, B[L], dst, (L == 3) ? 0 : 1);
    }
}